// covpool_5609227288680
// MI455X (gfx1250) — compile-verified
//
#include <hip/hip_runtime.h>
#include <hip/hip_bf16.h>

// Covariance pooling: y[b] = (1/M) * X * Xc^T, X = x[b] in [DIM, M] (fp32).
// Centering only one operand is exact because Xc * 1 == 0.
// Compute-bound on the fp32 matrix pipe => V_WMMA_F32_16X16X4_F32 from LDS,
// with 4x4 register tile blocking per wave: 16 WMMAs per K-step from
// 4 A-frags + 4 B-frags (4 B/lane of LDS fragment traffic per WMMA).

typedef __attribute__((ext_vector_type(2))) float v2f;
typedef __attribute__((ext_vector_type(8))) float v8f;

#define DIM  512        // channels
#define MLEN 256        // h*w
#define BTM  256        // output panel rows per workgroup
#define BTN  128        // output panel cols per workgroup
#define KC   64         // K chunk staged in LDS
#define LDA  (KC + 4)   // pad 4 floats: row stride = 4 dwords mod 64 banks -> conflict-free

__global__ __launch_bounds__(256)
void covpool_wmma_f32(const float* __restrict__ x, float* __restrict__ y) {
    __shared__ float sA[BTM * LDA];      // raw rows     i0 .. i0+255
    __shared__ float sB[BTN * LDA];      // centered rows j0 .. j0+127
    __shared__ float sMeanPart[2 * BTN];
    __shared__ float sMean[BTN];

    const int b    = blockIdx.z;
    const int i0   = blockIdx.y * BTM;   // output row panel
    const int j0   = blockIdx.x * BTN;   // output col panel
    const int tid  = threadIdx.x;
    const int lane = tid & 31;
    const int wave = tid >> 5;           // 0..7

    const float* __restrict__ xb = x + (size_t)b * DIM * MLEN;

    // ---- Pass 1: row means of the B-side panel (rows j0..j0+127) ----
    {
        const int row = tid & (BTN - 1); // 0..127
        const int seg = tid >> 7;        // 0..1, 128 elements each
        const float* p = xb + (size_t)(j0 + row) * MLEN + seg * 128;
        float s = 0.0f;
        #pragma unroll
        for (int k = 0; k < 128; k += 4) {
            float4 v = *(const float4*)(p + k);
            s += (v.x + v.y) + (v.z + v.w);
        }
        sMeanPart[seg * BTN + row] = s;
    }
    __syncthreads();
    if (tid < BTN) {
        sMean[tid] = (sMeanPart[tid] + sMeanPart[BTN + tid]) * (1.0f / MLEN);
    }
    __syncthreads();

    // Wave tile block: 4 row-tiles x 4 col-tiles of 16x16 (64x64 output).
    const int wr = wave >> 1;            // 0..3 -> rows wr*64 ..
    const int wc = wave & 1;             // 0..1 -> cols wc*64 ..

    v8f acc[4][4] = {};

    const int mrow = lane & 15;          // A: M row / B: N col within a tile
    const int kofs = (lane >> 4) * 2;    // K sub-offset per half-wave

    for (int kk = 0; kk < MLEN; kk += KC) {
        // ---- Stage A (raw, 256 rows) and B (centered, 128 rows) into LDS ----
        {
            const int rr = tid >> 1;     // 0..127
            const int c4 = tid & 1;      // 0..1
            // A: two row-halves of 128 rows each
            #pragma unroll
            for (int rh = 0; rh < 2; ++rh) {
                const int r = rh * 128 + rr;
                const float* srcA = xb + (size_t)(i0 + r) * MLEN + kk;
                float* dstA = sA + r * LDA;
                #pragma unroll
                for (int u = 0; u < 8; ++u) {
                    const int c = (c4 + u * 2) * 4;  // 0..60 step 4 (dwords)
                    *(float4*)(dstA + c) = *(const float4*)(srcA + c);
                }
            }
            // B: 128 rows, centered
            {
                const float* srcB = xb + (size_t)(j0 + rr) * MLEN + kk;
                float* dstB = sB + rr * LDA;
                const float mu = sMean[rr];
                #pragma unroll
                for (int u = 0; u < 8; ++u) {
                    const int c = (c4 + u * 2) * 4;
                    float4 v = *(const float4*)(srcB + c);
                    v.x -= mu; v.y -= mu; v.z -= mu; v.w -= mu;
                    *(float4*)(dstB + c) = v;
                }
            }
        }
        __syncthreads();

        // Fragment base pointers (even float offsets -> 8B aligned for v2f).
        const float* aBase = sA + (wr * 64 + mrow) * LDA + kofs;
        const float* bBase = sB + (wc * 64 + mrow) * LDA + kofs;

        #pragma unroll
        for (int k0 = 0; k0 < KC; k0 += 4) {
            v2f a0 = *(const v2f*)(aBase + k0);
            v2f a1 = *(const v2f*)(aBase + 16 * LDA + k0);
            v2f a2 = *(const v2f*)(aBase + 32 * LDA + k0);
            v2f a3 = *(const v2f*)(aBase + 48 * LDA + k0);
            v2f b0 = *(const v2f*)(bBase + k0);
            v2f b1 = *(const v2f*)(bBase + 16 * LDA + k0);
            v2f b2 = *(const v2f*)(bBase + 32 * LDA + k0);
            v2f b3 = *(const v2f*)(bBase + 48 * LDA + k0);
            // 16 WMMAs reuse 4 A frags + 4 B frags.
            acc[0][0] = __builtin_amdgcn_wmma_f32_16x16x4_f32(false, a0, false, b0, (short)0, acc[0][0], false, false);
            acc[0][1] = __builtin_amdgcn_wmma_f32_16x16x4_f32(false, a0, false, b1, (short)0, acc[0][1], false, false);
            acc[0][2] = __builtin_amdgcn_wmma_f32_16x16x4_f32(false, a0, false, b2, (short)0, acc[0][2], false, false);
            acc[0][3] = __builtin_amdgcn_wmma_f32_16x16x4_f32(false, a0, false, b3, (short)0, acc[0][3], false, false);
            acc[1][0] = __builtin_amdgcn_wmma_f32_16x16x4_f32(false, a1, false, b0, (short)0, acc[1][0], false, false);
            acc[1][1] = __builtin_amdgcn_wmma_f32_16x16x4_f32(false, a1, false, b1, (short)0, acc[1][1], false, false);
            acc[1][2] = __builtin_amdgcn_wmma_f32_16x16x4_f32(false, a1, false, b2, (short)0, acc[1][2], false, false);
            acc[1][3] = __builtin_amdgcn_wmma_f32_16x16x4_f32(false, a1, false, b3, (short)0, acc[1][3], false, false);
            acc[2][0] = __builtin_amdgcn_wmma_f32_16x16x4_f32(false, a2, false, b0, (short)0, acc[2][0], false, false);
            acc[2][1] = __builtin_amdgcn_wmma_f32_16x16x4_f32(false, a2, false, b1, (short)0, acc[2][1], false, false);
            acc[2][2] = __builtin_amdgcn_wmma_f32_16x16x4_f32(false, a2, false, b2, (short)0, acc[2][2], false, false);
            acc[2][3] = __builtin_amdgcn_wmma_f32_16x16x4_f32(false, a2, false, b3, (short)0, acc[2][3], false, false);
            acc[3][0] = __builtin_amdgcn_wmma_f32_16x16x4_f32(false, a3, false, b0, (short)0, acc[3][0], false, false);
            acc[3][1] = __builtin_amdgcn_wmma_f32_16x16x4_f32(false, a3, false, b1, (short)0, acc[3][1], false, false);
            acc[3][2] = __builtin_amdgcn_wmma_f32_16x16x4_f32(false, a3, false, b2, (short)0, acc[3][2], false, false);
            acc[3][3] = __builtin_amdgcn_wmma_f32_16x16x4_f32(false, a3, false, b3, (short)0, acc[3][3], false, false);
        }
        __syncthreads();
    }

    // ---- Epilogue: scale by 1/M and store ----
    // C/D layout: VGPR r -> lanes 0-15: row M=r, col=lane; lanes 16-31: row M=r+8.
    const float scale = 1.0f / MLEN;
    const int ncol = lane & 15;
    const int rsel = (lane >> 4) * 8;
    float* __restrict__ yb = y + (size_t)b * DIM * DIM;
    #pragma unroll
    for (int am = 0; am < 4; ++am) {
        #pragma unroll
        for (int bn = 0; bn < 4; ++bn) {
            const int col = j0 + wc * 64 + bn * 16 + ncol;
            #pragma unroll
            for (int r = 0; r < 8; ++r) {
                const int row = i0 + wr * 64 + am * 16 + r + rsel;
                yb[(size_t)row * DIM + col] = acc[am][bn][r] * scale;
            }
        }
    }
}

extern "C" void kernel_launch(void* const* d_in, const int* in_sizes, int n_in,
                              void* d_out, int out_size, void* d_ws, size_t ws_size,
                              hipStream_t stream) {
    (void)in_sizes; (void)n_in; (void)d_ws; (void)ws_size; (void)out_size;
    const float* x = (const float*)d_in[0];   // [64, 512, 16, 16] fp32
    float* y = (float*)d_out;                 // [64, 512, 512] fp32

    dim3 grid(DIM / BTN, DIM / BTM, 64);      // (4, 2, 64)
    dim3 block(256);
    covpool_wmma_f32<<<grid, block, 0, stream>>>(x, y);
}